// DetectionTargetLayer_65566970740958
// MI455X (gfx1250) — compile-verified
//
#include <hip/hip_runtime.h>
#include <math.h>

#define BB 8
#define NN 2000
#define GG 100
#define HH 320
#define WW 320
#define TROIS 200
#define NUM_POS 66
#define NUM_NEG 134
#define MH 28
#define MW 28
#define EPSF 1e-8f

#define CLS_OFF   (BB*TROIS*4)                 // 6400
#define DELTA_OFF (CLS_OFF + BB*TROIS)         // 8000
#define MASK_OFF  (DELTA_OFF + BB*TROIS*4)     // 14400

typedef __attribute__((ext_vector_type(2))) float v2f;
typedef __attribute__((ext_vector_type(8))) float v8f;

__device__ __forceinline__ int imin(int a, int b) { return a < b ? a : b; }
__device__ __forceinline__ int imax(int a, int b) { return a > b ? a : b; }

// ---------------- Kernel 1: IoU max / argmax per proposal ----------------
__global__ void k_iou(const float* __restrict__ prop, const float* __restrict__ gtb,
                      float* __restrict__ ioumax, int* __restrict__ gmax,
                      int* __restrict__ posf, int* __restrict__ negf) {
    __shared__ float sg[GG][4];
    __shared__ int   sgv[GG];
    const int b = blockIdx.y;
    const int n = blockIdx.x * blockDim.x + threadIdx.x;

    for (int g = threadIdx.x; g < GG; g += blockDim.x) {
        const float* p = gtb + (size_t)(b * GG + g) * 4;
        float g0 = p[0], g1 = p[1], g2 = p[2], g3 = p[3];
        sg[g][0] = g0; sg[g][1] = g1; sg[g][2] = g2; sg[g][3] = g3;
        sgv[g] = (fabsf(g0) > 0.f || fabsf(g1) > 0.f || fabsf(g2) > 0.f || fabsf(g3) > 0.f) ? 1 : 0;
    }
    __syncthreads();
    if (n >= NN) return;

    const float* pp = prop + (size_t)(b * NN + n) * 4;
    float a0 = pp[0], a1 = pp[1], a2 = pp[2], a3 = pp[3];
    int vp = (fabsf(a0) > 0.f || fabsf(a1) > 0.f || fabsf(a2) > 0.f || fabsf(a3) > 0.f) ? 1 : 0;
    float area_a = (a2 - a0) * (a3 - a1);

    float best = -1e30f; int bi = 0;
    for (int g = 0; g < GG; ++g) {
        float v = -1.0f;
        if (vp && sgv[g]) {
            float y1 = fmaxf(a0, sg[g][0]);
            float x1 = fmaxf(a1, sg[g][1]);
            float y2 = fminf(a2, sg[g][2]);
            float x2 = fminf(a3, sg[g][3]);
            float inter = fmaxf(y2 - y1, 0.f) * fmaxf(x2 - x1, 0.f);
            float area_b = (sg[g][2] - sg[g][0]) * (sg[g][3] - sg[g][1]);
            float uni = area_a + area_b - inter;
            v = inter / fmaxf(uni, EPSF);
        }
        if (v > best) { best = v; bi = g; }   // first-max tie-break == jnp.argmax
    }
    int o = b * NN + n;
    ioumax[o] = best;
    gmax[o]   = bi;
    posf[o]   = (best >= 0.5f) ? 1 : 0;
    negf[o]   = ((best < 0.5f) && vp) ? 1 : 0;
}

// ---------------- Kernel 2: stable top-k selection per batch ----------------
__global__ void k_select(const float* __restrict__ ioumax, const int* __restrict__ posf,
                         const int* __restrict__ negf, int* __restrict__ pos_idx,
                         int* __restrict__ neg_idx, int* __restrict__ counts) {
    __shared__ float key[NN];
    __shared__ float rv[256];
    __shared__ int   ri[256];
    __shared__ int   rc[256];
    const int b = blockIdx.x;
    const int t = threadIdx.x;

    // counts (pack pos | neg<<16)
    int cp = 0, cn = 0;
    for (int n = t; n < NN; n += 256) { cp += posf[b*NN+n]; cn += negf[b*NN+n]; }
    rc[t] = cp | (cn << 16);
    __syncthreads();
    for (int s = 128; s > 0; s >>= 1) {
        if (t < s) rc[t] += rc[t + s];
        __syncthreads();
    }
    if (t == 0) {
        int pos_total = rc[0] & 0xFFFF;
        int neg_total = rc[0] >> 16;
        int pc = imin(pos_total, NUM_POS);
        int r32 = (int)((float)pc / 0.33f);           // matches astype(int32) truncation
        int ncnt = imin(imin(r32 - pc, neg_total), NUM_NEG);
        counts[2*b]   = pc;
        counts[2*b+1] = ncnt;
    }

    // positives: key = pos ? max : -1
    __syncthreads();
    for (int n = t; n < NN; n += 256) key[n] = posf[b*NN+n] ? ioumax[b*NN+n] : -1.0f;
    __syncthreads();
    for (int r = 0; r < NUM_POS; ++r) {
        float bv = -1e30f; int bi = NN;
        for (int n = t; n < NN; n += 256) { float v = key[n]; if (v > bv) { bv = v; bi = n; } }
        rv[t] = bv; ri[t] = bi;
        __syncthreads();
        for (int s = 128; s > 0; s >>= 1) {
            if (t < s) {
                float ov = rv[t+s]; int oi = ri[t+s];
                if (ov > rv[t] || (ov == rv[t] && oi < ri[t])) { rv[t] = ov; ri[t] = oi; }
            }
            __syncthreads();
        }
        if (t == 0) { pos_idx[b*NUM_POS + r] = ri[0]; key[ri[0]] = -3e38f; }
        __syncthreads();
    }

    // negatives: key = neg ? max : -2
    for (int n = t; n < NN; n += 256) key[n] = negf[b*NN+n] ? ioumax[b*NN+n] : -2.0f;
    __syncthreads();
    for (int r = 0; r < NUM_NEG; ++r) {
        float bv = -1e30f; int bi = NN;
        for (int n = t; n < NN; n += 256) { float v = key[n]; if (v > bv) { bv = v; bi = n; } }
        rv[t] = bv; ri[t] = bi;
        __syncthreads();
        for (int s = 128; s > 0; s >>= 1) {
            if (t < s) {
                float ov = rv[t+s]; int oi = ri[t+s];
                if (ov > rv[t] || (ov == rv[t] && oi < ri[t])) { rv[t] = ov; ri[t] = oi; }
            }
            __syncthreads();
        }
        if (t == 0) { neg_idx[b*NUM_NEG + r] = ri[0]; key[ri[0]] = -3e38f; }
        __syncthreads();
    }
}

// ---------------- Kernel 3: gather rois / class / deltas ----------------
__global__ void k_gather(const float* __restrict__ prop, const int* __restrict__ cls,
                         const float* __restrict__ gtb, const int* __restrict__ gmax,
                         const int* __restrict__ pos_idx, const int* __restrict__ neg_idx,
                         const int* __restrict__ counts, float* __restrict__ out,
                         float* __restrict__ crop_box, int* __restrict__ crop_g) {
    const int b = blockIdx.x;
    const int i = threadIdx.x;
    if (i >= TROIS) return;
    const int pc = counts[2*b];
    const int nc = counts[2*b+1];

    float roi[4] = {0.f, 0.f, 0.f, 0.f};
    float del[4] = {0.f, 0.f, 0.f, 0.f};
    float cv = 0.f;

    if (i < NUM_POS) {
        int ok = (i < pc);
        int g = 0;
        if (ok) {
            int idx = pos_idx[b*NUM_POS + i];
            const float* p = prop + (size_t)(b*NN + idx) * 4;
            roi[0] = p[0]; roi[1] = p[1]; roi[2] = p[2]; roi[3] = p[3];
            g = gmax[b*NN + idx];
            const float* gt = gtb + (size_t)(b*GG + g) * 4;
            cv = (float)cls[b*GG + g];
            float h  = fmaxf(roi[2] - roi[0], EPSF);
            float w  = fmaxf(roi[3] - roi[1], EPSF);
            float cy = roi[0] + 0.5f * h;
            float cx = roi[1] + 0.5f * w;
            float gh = fmaxf(gt[2] - gt[0], EPSF);
            float gw = fmaxf(gt[3] - gt[1], EPSF);
            float gcy = gt[0] + 0.5f * gh;
            float gcx = gt[1] + 0.5f * gw;
            del[0] = ((gcy - cy) / h) / 0.1f;
            del[1] = ((gcx - cx) / w) / 0.1f;
            del[2] = logf(gh / h) / 0.2f;
            del[3] = logf(gw / w) / 0.2f;
        }
        float* cb = crop_box + (size_t)(b*NUM_POS + i) * 4;
        cb[0] = roi[0]; cb[1] = roi[1]; cb[2] = roi[2]; cb[3] = roi[3];
        crop_g[b*NUM_POS + i] = g;
    } else {
        int k = i - NUM_POS;
        if (k < nc) {
            int idx = neg_idx[b*NUM_NEG + k];
            const float* p = prop + (size_t)(b*NN + idx) * 4;
            roi[0] = p[0]; roi[1] = p[1]; roi[2] = p[2]; roi[3] = p[3];
        }
    }
    const int slot = b * TROIS + i;
    float* ro = out + (size_t)slot * 4;
    ro[0] = roi[0]; ro[1] = roi[1]; ro[2] = roi[2]; ro[3] = roi[3];
    out[CLS_OFF + slot] = cv;
    float* dl = out + DELTA_OFF + (size_t)slot * 4;
    dl[0] = del[0]; dl[1] = del[1]; dl[2] = del[2]; dl[3] = del[3];
}

// ---------------- Kernel 4: mask crop-resize via f32 WMMA ----------------
// out(32x32,use 28x28) = Wy(32x320) * tmp(320x32), K split into two 160 passes.
#define KC 160
__global__ __launch_bounds__(128) void k_mask(const unsigned char* __restrict__ masks,
                                              const float* __restrict__ crop_box,
                                              const int* __restrict__ crop_g,
                                              const int* __restrict__ counts,
                                              float* __restrict__ out) {
    const int blk = blockIdx.x;               // b*TROIS + s
    const int b = blk / TROIS;
    const int s = blk % TROIS;
    float* o = out + MASK_OFF + (size_t)blk * (MH * MW);
    const int t = threadIdx.x;

    bool doCrop = (s < NUM_POS) && (s < counts[2*b]);
    if (!doCrop) {
        for (int e = t; e < MH * MW; e += 128) o[e] = 0.f;
        return;
    }

    __shared__ __align__(16) float sWy[32 * KC];
    __shared__ __align__(16) float sTmp[32 * KC];
    __shared__ int   sX0[MW];
    __shared__ float sWx[MW];
    __shared__ int   sY0[MH];
    __shared__ float sWyv[MH];

    const float* box = crop_box + (size_t)(b*NUM_POS + s) * 4;
    const float y1 = box[0], x1 = box[1], y2 = box[2], x2 = box[3];
    const int g = crop_g[b*NUM_POS + s];
    const unsigned char* mb = masks + (size_t)b * HH * WW * GG + g;
    __builtin_prefetch(mb, 0, 0);

    if (t < MH) {
        float f = (float)t / 27.0f;
        float ys = y1 * 319.0f + f * (y2 - y1) * 319.0f;
        ys = fminf(fmaxf(ys, 0.0f), 319.0f);
        int yy0 = imin(imax((int)floorf(ys), 0), 318);
        sY0[t] = yy0; sWyv[t] = ys - (float)yy0;
        float xs = x1 * 319.0f + f * (x2 - x1) * 319.0f;
        xs = fminf(fmaxf(xs, 0.0f), 319.0f);
        int xx0 = imin(imax((int)floorf(xs), 0), 318);
        sX0[t] = xx0; sWx[t] = xs - (float)xx0;
    }
    __syncthreads();

    const int lane = t & 31;
    const int wave = t >> 5;
    const int half = lane >> 4;
    const int l    = lane & 15;
    const int Mt   = wave >> 1;
    const int Nt   = wave & 1;

    v8f c = {0.f, 0.f, 0.f, 0.f, 0.f, 0.f, 0.f, 0.f};

    for (int pass = 0; pass < 2; ++pass) {
        const int ybase = pass * KC;
        // zero Wy tile + tmp padding rows (28..31)
        for (int e = t; e < 32 * KC; e += 128) sWy[e] = 0.f;
        for (int e = t; e < 4 * KC; e += 128)  sTmp[28 * KC + e] = 0.f;
        __syncthreads();
        // scatter the 2-tap y-interpolation weights
        if (t < MH) {
            int yy = sY0[t]; float w = sWyv[t];
            if (yy >= ybase && yy < ybase + KC)         sWy[t * KC + (yy     - ybase)] = 1.0f - w;
            if (yy + 1 >= ybase && yy + 1 < ybase + KC) sWy[t * KC + (yy + 1 - ybase)] = w;
        }
        // x-interpolated mask columns: tmpT[j][y] (B matrix, N-major rows)
        for (int e = t; e < MW * KC; e += 128) {
            int j  = e / KC;
            int yo = e - j * KC;
            int y  = ybase + yo;
            int x0 = sX0[j];
            float wx = sWx[j];
            const unsigned char* row = mb + ((size_t)y * WW + x0) * GG;
            float m0 = (float)row[0];
            float m1 = (float)row[GG];
            sTmp[j * KC + yo] = m0 * (1.0f - wx) + m1 * wx;
        }
        __syncthreads();

        // D = A(16x4 f32) x B(4x16 f32) + C, K swept in steps of 4
        const float* aptr = &sWy[(Mt * 16 + l) * KC + 2 * half];
        const float* bptr = &sTmp[(Nt * 16 + l) * KC + 2 * half];
        for (int kb = 0; kb < KC; kb += 4) {
            v2f a  = *(const v2f*)(aptr + kb);
            v2f bv = *(const v2f*)(bptr + kb);
            c = __builtin_amdgcn_wmma_f32_16x16x4_f32(false, a, false, bv,
                                                      (short)0, c, false, false);
        }
        __syncthreads();   // protect LDS before next pass overwrites
    }

    // C/D layout: VGPR r -> (M = r + 8*half, N = lane%16) within the 16x16 tile
    #pragma unroll
    for (int r = 0; r < 8; ++r) {
        int row = Mt * 16 + half * 8 + r;
        int col = Nt * 16 + l;
        if (row < MH && col < MW) o[row * MW + col] = rintf(c[r]);  // round half-to-even
    }
}

extern "C" void kernel_launch(void* const* d_in, const int* in_sizes, int n_in,
                              void* d_out, int out_size, void* d_ws, size_t ws_size,
                              hipStream_t stream) {
    const float*         prop  = (const float*)d_in[0];
    const int*           cls   = (const int*)d_in[1];
    const float*         gtb   = (const float*)d_in[2];
    const unsigned char* masks = (const unsigned char*)d_in[3];  // jax bool = 1 byte
    float* out = (float*)d_out;

    float* ioumax  = (float*)d_ws;                       // B*N
    int*   gmax    = (int*)(ioumax + BB*NN);             // B*N
    int*   posf    = gmax + BB*NN;                       // B*N
    int*   negf    = posf + BB*NN;                       // B*N
    int*   pos_idx = negf + BB*NN;                       // B*NUM_POS
    int*   neg_idx = pos_idx + BB*NUM_POS;               // B*NUM_NEG
    int*   counts  = neg_idx + BB*NUM_NEG;               // B*2
    float* crop_box = (float*)(counts + 2*BB);           // B*NUM_POS*4
    int*   crop_g   = (int*)(crop_box + BB*NUM_POS*4);   // B*NUM_POS

    k_iou<<<dim3((NN + 255) / 256, BB), 256, 0, stream>>>(prop, gtb, ioumax, gmax, posf, negf);
    k_select<<<BB, 256, 0, stream>>>(ioumax, posf, negf, pos_idx, neg_idx, counts);
    k_gather<<<BB, 256, 0, stream>>>(prop, cls, gtb, gmax, pos_idx, neg_idx, counts,
                                     out, crop_box, crop_g);
    k_mask<<<BB * TROIS, 128, 0, stream>>>(masks, crop_box, crop_g, counts, out);
}